// MultiHeadAttention_4913442586758
// MI455X (gfx1250) — compile-verified
//
#include <hip/hip_runtime.h>
#include <math.h>

typedef __attribute__((ext_vector_type(2))) float v2f;
typedef __attribute__((ext_vector_type(8))) float v8f;

#define B_  16
#define C_  2048
#define H_  2
#define HS_ 64
#define CT_ (C_ / 16)            // 128 c-tiles per batch row
#define LOG2E 1.4426950408889634f

// lane <-> lane^16 exchange, immediate-offset ds_swizzle (group-of-32: xor=16, and=31)
__device__ __forceinline__ float swz_xor16(float v) {
    return __int_as_float(__builtin_amdgcn_ds_swizzle(__float_as_int(v), 0x401F));
}
__device__ __forceinline__ float fexp2(float x) { return __builtin_amdgcn_exp2f(x); }

// One wave per (batch, 16-query c-tile). Rank-2 attention:
//   s[c,f] = (xp[c] M_h) . xp[f],  M_h = Wq_h Wk_h^T * HS^-0.5 (2x2, log2e-folded)
//   out_h[c] = sum_f softmax_f(s)[c,f] * w_h[f],  w_h[f] = xp[f] . (Wv_h Wo_h)
// WMMA computes the TRANSPOSED tile S'[f][c] (A = keys, B = query coeffs) so the
// softmax axis f lies along the 8 D-VGPRs + one lane-pair: 3 ds_swizzles per tile.
__global__ __launch_bounds__(32) void mha_rank2_wmma_kernel(
    const float* __restrict__ x,
    const float* __restrict__ Wq,
    const float* __restrict__ Wk,
    const float* __restrict__ Wv,
    const float* __restrict__ Wo,
    const float* __restrict__ Wboth,
    float* __restrict__ out)
{
    __shared__ float s_xp[C_ * 2];   // xp = x + positional encoding   (16 KB)
    __shared__ float s_w[H_][C_];    // per-head value scalars w_h[f]  (16 KB)

    const int blk  = blockIdx.x;             // 0 .. B_*CT_-1
    const int b    = blk / CT_;
    const int t    = blk % CT_;
    const int c0   = t * 16;
    const int lane = threadIdx.x;            // wave32
    const int half = lane >> 4;
    const int nl   = lane & 15;
    const int nf   = c0 + 16;                // #keys this tile block needs

    // ---- 12 lane-parallel dot products over HS: M_h (2x2) and u_h (2) per head ----
    float r[12];
    {
        const int e0 = lane, e1 = lane + 32;
        #pragma unroll
        for (int h = 0; h < H_; ++h) {
            const float* wq = Wq + h * 2 * HS_;
            const float* wk = Wk + h * 2 * HS_;
            const float* wv = Wv + h * 2 * HS_;
            const float* wo = Wo + h * HS_;
            float q0a = wq[e0],       q0b = wq[e1];
            float q1a = wq[HS_ + e0], q1b = wq[HS_ + e1];
            float k0a = wk[e0],       k0b = wk[e1];
            float k1a = wk[HS_ + e0], k1b = wk[HS_ + e1];
            float v0a = wv[e0],       v0b = wv[e1];
            float v1a = wv[HS_ + e0], v1b = wv[HS_ + e1];
            float oa  = wo[e0],       ob  = wo[e1];
            r[h*6+0] = q0a*k0a + q0b*k0b;   // M00
            r[h*6+1] = q0a*k1a + q0b*k1b;   // M01
            r[h*6+2] = q1a*k0a + q1b*k0b;   // M10
            r[h*6+3] = q1a*k1a + q1b*k1b;   // M11
            r[h*6+4] = v0a*oa  + v0b*ob;    // u0
            r[h*6+5] = v1a*oa  + v1b*ob;    // u1
        }
        #pragma unroll
        for (int m = 16; m >= 1; m >>= 1) {
            #pragma unroll
            for (int k = 0; k < 12; ++k) r[k] += __shfl_xor(r[k], m);
        }
    }

    // ---- stage xp[f] and w_h[f] for f in [0, nf) into LDS (hot loop is LDS-only) ----
    const float* xb = x + (size_t)b * C_ * 2;
    for (int base = 0; base < nf; base += 32) {
        const int i = base + lane;
        if (i < nf) {
            const float fp  = (float)i;
            const float xp0 = xb[i*2+0] + sinf(fp);
            const float xp1 = xb[i*2+1] + cosf(fp);
            s_xp[i*2+0] = xp0;
            s_xp[i*2+1] = xp1;
            s_w[0][i] = xp0 * r[4]  + xp1 * r[5];
            s_w[1][i] = xp0 * r[10] + xp1 * r[11];
        }
    }
    __syncthreads();

    const float xq0 = s_xp[(c0 + nl) * 2 + 0];
    const float xq1 = s_xp[(c0 + nl) * 2 + 1];

    const float msc = 0.125f * LOG2E;   // HS^-0.5 and log2(e) folded into the scores
    float hout0 = 0.0f, hout1 = 0.0f;

    #pragma unroll
    for (int h = 0; h < H_; ++h) {
        const float M00 = r[h*6+0] * msc, M01 = r[h*6+1] * msc;
        const float M10 = r[h*6+2] * msc, M11 = r[h*6+3] * msc;
        // B operand (4x16): K=0 row = aM[c], K=1 row = bM[c]; K=2,3 rows zero.
        const float aM = xq0 * M00 + xq1 * M10;
        const float bM = xq0 * M01 + xq1 * M11;
        v2f Bq;
        Bq.x = half ? 0.0f : aM;
        Bq.y = half ? 0.0f : bM;

        const float* wrow = &s_w[h][0];
        float run_m = -INFINITY, run_s = 0.0f, run_o = 0.0f;

        auto tile = [&](int f0, bool diag) {
            // A operand (16x4): M=f_local=nl, K=0,1 = xp[f]; lanes>=16 garbage is
            // harmless because B's K=2,3 rows are zero.
            const v2f A = *(const v2f*)&s_xp[(f0 + nl) * 2];
            v8f Cz = {};
            v8f D = __builtin_amdgcn_wmma_f32_16x16x4_f32(
                false, A, false, Bq, (short)0, Cz, false, false);

            const float4 wa = *(const float4*)&wrow[f0 + 8 * half];
            const float4 wb = *(const float4*)&wrow[f0 + 8 * half + 4];

            float s[8];
            #pragma unroll
            for (int v = 0; v < 8; ++v) {
                s[v] = D[v];                                   // S'[f0+v+8*half][c0+nl]
                if (diag && (v + 8 * half > nl)) s[v] = -3.0e38f;   // causal mask
            }
            float tm = fmaxf(fmaxf(fmaxf(s[0], s[1]), fmaxf(s[2], s[3])),
                             fmaxf(fmaxf(s[4], s[5]), fmaxf(s[6], s[7])));
            tm = fmaxf(tm, swz_xor16(tm));
            const float nm   = fmaxf(run_m, tm);
            const float corr = fexp2(run_m - nm);
            const float p0 = fexp2(s[0] - nm), p1 = fexp2(s[1] - nm);
            const float p2 = fexp2(s[2] - nm), p3 = fexp2(s[3] - nm);
            const float p4 = fexp2(s[4] - nm), p5 = fexp2(s[5] - nm);
            const float p6 = fexp2(s[6] - nm), p7 = fexp2(s[7] - nm);
            float ps = ((p0 + p1) + (p2 + p3)) + ((p4 + p5) + (p6 + p7));
            float pw = p0 * wa.x + p1 * wa.y + p2 * wa.z + p3 * wa.w
                     + p4 * wb.x + p5 * wb.y + p6 * wb.z + p7 * wb.w;
            ps += swz_xor16(ps);
            pw += swz_xor16(pw);
            run_s = run_s * corr + ps;
            run_o = run_o * corr + pw;
            run_m = nm;
        };

        for (int f0 = 0; f0 < c0; f0 += 16) tile(f0, false);  // unmasked tiles
        tile(c0, true);                                       // diagonal (masked) tile

        const float o = run_o / run_s;
        if (h == 0) hout0 = o; else hout1 = o;
    }

    // ---- combine heads with Wboth; lane nl (<16) owns query row c0+nl ----
    if (lane < 16) {
        const float Wb00 = Wboth[0], Wb01 = Wboth[1];
        const float Wb10 = Wboth[2], Wb11 = Wboth[3];
        float2 o;
        o.x = hout0 * Wb00 + hout1 * Wb10;
        o.y = hout0 * Wb01 + hout1 * Wb11;
        *(float2*)&out[((size_t)b * C_ + c0 + nl) * 2] = o;
    }
}

extern "C" void kernel_launch(void* const* d_in, const int* in_sizes, int n_in,
                              void* d_out, int out_size, void* d_ws, size_t ws_size,
                              hipStream_t stream) {
    const float* x  = (const float*)d_in[0];
    const float* Wq = (const float*)d_in[1];
    const float* Wk = (const float*)d_in[2];
    const float* Wv = (const float*)d_in[3];
    const float* Wo = (const float*)d_in[4];
    const float* Wb = (const float*)d_in[5];
    float* out = (float*)d_out;
    (void)in_sizes; (void)n_in; (void)out_size; (void)d_ws; (void)ws_size;

    dim3 grid(B_ * CT_);   // 2048 single-wave workgroups
    dim3 block(32);
    hipLaunchKernelGGL(mha_rank2_wmma_kernel, grid, block, 0, stream,
                       x, Wq, Wk, Wv, Wo, Wb, out);
}